// PointUpsampleAttn_5549097746755
// MI455X (gfx1250) — compile-verified
//
#include <hip/hip_runtime.h>

typedef float v2f __attribute__((ext_vector_type(2)));
typedef float v8f __attribute__((ext_vector_type(8)));

#define B_    4
#define N_    16384
#define S_    4096
#define C_    256
#define QPB   128    // queries per block: 8 waves x 16
#define WAVES 8

// Branchless insertion of (x, n) into sorted triple d0<=d1<=d2.
// 3 compares + cndmask chains; no exec-mask branches.
__device__ __forceinline__ void insert3(float x, int n,
                                        float& d0, float& d1, float& d2,
                                        int& i0, int& i1, int& i2)
{
    const bool c0 = x < d0;
    const bool c1 = x < d1;
    const bool c2 = x < d2;
    const float nd2 = c1 ? d1 : (c2 ? x : d2);
    const int   ni2 = c1 ? i1 : (c2 ? n : i2);
    const float nd1 = c0 ? d0 : (c1 ? x : d1);
    const int   ni1 = c0 ? i0 : (c1 ? n : i1);
    const float nd0 = c0 ? x  : d0;
    const int   ni0 = c0 ? n  : i0;
    d0 = nd0; d1 = nd1; d2 = nd2;
    i0 = ni0; i1 = ni1; i2 = ni2;
}

__global__ __launch_bounds__(256) void knn3_interp_wmma(
    const float* __restrict__ q,
    const float* __restrict__ kp,
    const float* __restrict__ v,
    float* __restrict__ out)
{
    __shared__ float mrg_d[WAVES][16][16][3];   // 24 KB
    __shared__ int   mrg_i[WAVES][16][16][3];   // 24 KB
    __shared__ float s_w[QPB][3];               // 1.5 KB
    __shared__ int   s_i[QPB][3];               // 1.5 KB

    const int tid  = threadIdx.x;
    const int wave = tid >> 5;
    const int lane = tid & 31;
    const int half = lane >> 4;      // 0: lanes 0-15, 1: lanes 16-31
    const int lm   = lane & 15;      // M index for A / N index for B,C

    const int blocksPerBatch = N_ / QPB;
    const int b      = blockIdx.x / blocksPerBatch;
    const int n_base = (blockIdx.x % blocksPerBatch) * QPB;
    const int wave_n = n_base + wave * 16;

    // ---- A fragment: A[m][k] = -2*q_coord (k=0..2), 0 for k=3.
    // f32 16x4 A layout: lanes 0-15 -> VGPR0=K0, VGPR1=K1; lanes 16-31 -> K2, K3.
    const int nq = wave_n + lm;
    const float qx = q[((size_t)b * N_ + nq) * 3 + 0];
    const float qy = q[((size_t)b * N_ + nq) * 3 + 1];
    const float qz = q[((size_t)b * N_ + nq) * 3 + 2];
    v2f a;
    a.x = -2.0f * (half ? qz : qx);
    a.y = half ? 0.0f : (-2.0f * qy);

    // ---- Running per-lane top-3 (smallest score) for each of 8 C rows.
    float bd[8][3];
    int   bi[8][3];
#pragma unroll
    for (int r = 0; r < 8; ++r)
#pragma unroll
        for (int j = 0; j < 3; ++j) { bd[r][j] = 3.0e38f; bi[r][j] = 0; }

    const float* kb = kp + (size_t)b * S_ * 3;
    // 32 keys per trip: two independent WMMA tiles to overlap load/WMMA/VALU.
    for (int s0 = 0; s0 < S_; s0 += 32) {
        const int nA = s0 + lm;            // tile A key for this lane
        const int nB = s0 + 16 + lm;       // tile B key for this lane

        const float kx0 = kb[nA * 3 + 0];
        const float ky0 = kb[nA * 3 + 1];
        const float kz0 = kb[nA * 3 + 2];
        const float kx1 = kb[nB * 3 + 0];
        const float ky1 = kb[nB * 3 + 1];
        const float kz1 = kb[nB * 3 + 2];

        const float kk0 = kx0 * kx0 + ky0 * ky0 + kz0 * kz0;
        const float kk1 = kx1 * kx1 + ky1 * ky1 + kz1 * kz1;

        // B fragments (4x16): lanes 0-15 -> K0=x, K1=y; lanes 16-31 -> K2=z, K3=0.
        v2f bf0, bf1;
        bf0.x = half ? kz0 : kx0;
        bf0.y = half ? 0.0f : ky0;
        bf1.x = half ? kz1 : kx1;
        bf1.y = half ? 0.0f : ky1;

        // C pre-load: C[m][n] = |k_n|^2 (same N for all 8 row-VGPRs per lane).
        v8f c0, c1;
#pragma unroll
        for (int j = 0; j < 8; ++j) { c0[j] = kk0; c1[j] = kk1; }

        // D[m][n] = |k_n|^2 - 2*q_m . k_n  (exact f32 WMMA)
        c0 = __builtin_amdgcn_wmma_f32_16x16x4_f32(false, a, false, bf0,
                                                   (short)0, c0, false, false);
        c1 = __builtin_amdgcn_wmma_f32_16x16x4_f32(false, a, false, bf1,
                                                   (short)0, c1, false, false);

#pragma unroll
        for (int r = 0; r < 8; ++r)
            insert3(c0[r], nA, bd[r][0], bd[r][1], bd[r][2],
                             bi[r][0], bi[r][1], bi[r][2]);
#pragma unroll
        for (int r = 0; r < 8; ++r)
            insert3(c1[r], nB, bd[r][0], bd[r][1], bd[r][2],
                             bi[r][0], bi[r][1], bi[r][2]);
    }

    // ---- Dump per-lane candidates: C-row r, this half -> query m = r + 8*half.
#pragma unroll
    for (int r = 0; r < 8; ++r) {
        const int qm = r + half * 8;
#pragma unroll
        for (int j = 0; j < 3; ++j) {
            mrg_d[wave][qm][lm][j] = bd[r][j];
            mrg_i[wave][qm][lm][j] = bi[r][j];
        }
    }
    __syncthreads();

    // ---- Merge 16 lanes x 3 candidates -> global top-3 for query m (lane m).
    if (lane < 16) {
        const int m = lane;                 // lane m loaded query m's coords above
        float d0 = 3.0e38f, d1 = 3.0e38f, d2 = 3.0e38f;
        int   i0 = 0, i1 = 0, i2 = 0;
        for (int slot = 0; slot < 16; ++slot) {
#pragma unroll
            for (int j = 0; j < 3; ++j) {
                insert3(mrg_d[wave][m][slot][j], mrg_i[wave][m][slot][j],
                        d0, d1, d2, i0, i1, i2);
            }
        }
        const float qq = qx * qx + qy * qy + qz * qz;
        d0 += qq; d1 += qq; d2 += qq;       // true squared distances
        const float r0 = 1.0f / (d0 + 1e-8f);
        const float r1 = 1.0f / (d1 + 1e-8f);
        const float r2 = 1.0f / (d2 + 1e-8f);
        const float inv = 1.0f / (r0 + r1 + r2);
        const int lq = wave * 16 + m;
        s_w[lq][0] = r0 * inv; s_w[lq][1] = r1 * inv; s_w[lq][2] = r2 * inv;
        s_i[lq][0] = i0;       s_i[lq][1] = i1;       s_i[lq][2] = i2;
    }
    __syncthreads();

    // ---- Gather + weighted blend; out layout (B, C, N).
    // threads 0-127 share c, consecutive n -> 512B-contiguous store runs;
    // v rows (16 MB total) are L2-resident (192 MB L2).
    const int ln = tid & 127;
    const int n  = n_base + ln;
    const float w0 = s_w[ln][0], w1 = s_w[ln][1], w2 = s_w[ln][2];
    const float* p0 = v + ((size_t)b * S_ + s_i[ln][0]) * C_;
    const float* p1 = v + ((size_t)b * S_ + s_i[ln][1]) * C_;
    const float* p2 = v + ((size_t)b * S_ + s_i[ln][2]) * C_;
    int c = tid >> 7;                                    // 0 or 1
    float* op = out + ((size_t)b * C_ + c) * (size_t)N_ + n;
    for (int it = 0; it < C_ / 2; ++it, c += 2, op += 2 * (size_t)N_) {
        op[0] = w0 * p0[c] + w1 * p1[c] + w2 * p2[c];
    }
}

extern "C" void kernel_launch(void* const* d_in, const int* in_sizes, int n_in,
                              void* d_out, int out_size, void* d_ws, size_t ws_size,
                              hipStream_t stream) {
    const float* q  = (const float*)d_in[0];
    const float* kp = (const float*)d_in[1];
    const float* v  = (const float*)d_in[2];
    float* out = (float*)d_out;
    dim3 grid(B_ * (N_ / QPB));   // 512 blocks
    knn3_interp_wmma<<<grid, 256, 0, stream>>>(q, kp, v, out);
}